// Model_33414845563410
// MI455X (gfx1250) — compile-verified
//
#include <hip/hip_runtime.h>

// ---------------------------------------------------------------------------
// HMM 2x2 ordered matrix-product reduction for gfx1250 (MI455X).
// tokens: int32[4194304], ls: float32[128][2][2], out: float32[2]
// v = v0 @ (prod_t ls[tok[t]]) @ ls[tok[T-1]]  with v0 = [1,0]  => row 0.
// Bandwidth-bound: 16.8 MB of tokens @ 23.3 TB/s ~ 0.72 us floor.
// ---------------------------------------------------------------------------

#define T_TOTAL      4194304
#define NBLK         1024
#define TPB          256
#define TOK_PER_THR  16          // contiguous tokens per thread (kernel 1)
#define TILE_STRIDE  20          // dwords per thread row in LDS (16 + 4 pad, 16B aligned)

#if defined(__has_builtin)
# if __has_builtin(__builtin_amdgcn_global_load_async_to_lds_b128)
#  define HAVE_ASYNC_B128 1
# endif
# if __has_builtin(__builtin_amdgcn_s_wait_asynccnt)
#  define HAVE_WAIT_ASYNC 1
# endif
#endif

// Builtin parameter types (from hipcc diagnostic): AS1 / AS3 pointers to int4.
typedef int v4i __attribute__((ext_vector_type(4)));
typedef __attribute__((address_space(1))) v4i gv4i;   // global (AS1 == flat global addr)
typedef __attribute__((address_space(3))) v4i lv4i;   // LDS (AS3 == low 32 bits of flat LDS addr)

// 2x2 row-major product: a=(a00,a01,a10,a11), c = a * b
__device__ __forceinline__ float4 mm2(const float4 a, const float4 b) {
    float4 c;
    c.x = fmaf(a.y, b.z, a.x * b.x);
    c.y = fmaf(a.y, b.w, a.x * b.y);
    c.z = fmaf(a.w, b.z, a.z * b.x);
    c.w = fmaf(a.w, b.w, a.z * b.y);
    return c;
}

__global__ __launch_bounds__(TPB)
void hmm_partial(const int* __restrict__ tokens,
                 const float* __restrict__ ls,
                 float4* __restrict__ partial) {
    __shared__ __align__(16) float lsTab[512];                 // 2 KB: 128 matrices
    __shared__ __align__(16) int   tokTile[TPB * TILE_STRIDE]; // 20 KB padded token tile
    __shared__ __align__(16) float4 red[TPB];                  // 4 KB reduce array

    const int t = threadIdx.x;
    const int b = blockIdx.x;
    const long long blockTok = (long long)b * (TPB * TOK_PER_THR);   // 4096 tokens/block

    // Stage transition table into LDS (512 floats).
    lsTab[t]       = ls[t];
    lsTab[t + 256] = ls[t + 256];

    // Stage the block's 16 KB token range into LDS: 1024 b128 transfers,
    // coalesced global source, scattered to the padded per-thread layout.
#pragma unroll
    for (int i = 0; i < 4; ++i) {
        const int s     = i * TPB + t;     // int4 slot 0..1023 within block range
        const int owner = s >> 2;          // which thread's row
        const int k4    = s & 3;           // which int4 within the row
        const int* gsrc = tokens + blockTok + (long long)s * 4;
        int*       ldst = &tokTile[owner * TILE_STRIDE + k4 * 4];
#if HAVE_ASYNC_B128
        // gfx1250 async copy engine: global -> LDS, tracked by ASYNCcnt.
        __builtin_amdgcn_global_load_async_to_lds_b128(
            (gv4i*)(unsigned long long)(const void*)gsrc,
            (lv4i*)(unsigned)(unsigned long long)(void*)ldst,
            0, 0);
#else
        *(int4*)ldst = *(const int4*)gsrc;
#endif
    }
#if HAVE_ASYNC_B128
# if HAVE_WAIT_ASYNC
    __builtin_amdgcn_s_wait_asynccnt(0);
# else
    asm volatile("s_wait_asynccnt 0" ::: "memory");
# endif
#endif
    __syncthreads();

    // Chain 16 contiguous tokens: m = M[tok0] * M[tok1] * ... * M[tok15]
    const float4* lsT4  = (const float4*)lsTab;
    const int4*   myTok = (const int4*)&tokTile[t * TILE_STRIDE];

    int4 tt = myTok[0];
    float4 m = lsT4[tt.x];
    m = mm2(m, lsT4[tt.y]);
    m = mm2(m, lsT4[tt.z]);
    m = mm2(m, lsT4[tt.w]);
#pragma unroll
    for (int q = 1; q < 4; ++q) {
        tt = myTok[q];
        m = mm2(m, lsT4[tt.x]);
        m = mm2(m, lsT4[tt.y]);
        m = mm2(m, lsT4[tt.z]);
        m = mm2(m, lsT4[tt.w]);
    }

    // Ordered tree reduction 256 -> 1 (left segment * right segment).
    red[t] = m;
    __syncthreads();
#pragma unroll
    for (int s2 = 1; s2 < TPB; s2 <<= 1) {
        if ((t & (2 * s2 - 1)) == 0)
            red[t] = mm2(red[t], red[t + s2]);
        __syncthreads();
    }
    if (t == 0)
        partial[b] = red[0];
}

__global__ __launch_bounds__(TPB)
void hmm_final(const float4* __restrict__ partial,
               const int* __restrict__ tokens,
               const float* __restrict__ ls,
               float* __restrict__ out) {
    __shared__ __align__(16) float4 red[TPB];
    const int t = threadIdx.x;

    // 4 consecutive block-products per thread, in order.
    float4 m = partial[t * 4 + 0];
    m = mm2(m, partial[t * 4 + 1]);
    m = mm2(m, partial[t * 4 + 2]);
    m = mm2(m, partial[t * 4 + 3]);

    red[t] = m;
    __syncthreads();
#pragma unroll
    for (int s2 = 1; s2 < TPB; s2 <<= 1) {
        if ((t & (2 * s2 - 1)) == 0)
            red[t] = mm2(red[t], red[t + s2]);
        __syncthreads();
    }

    if (t == 0) {
        // Faithful to reference: last token's matrix applied a second time.
        const int j = tokens[T_TOTAL - 1];
        const float4 ml = ((const float4*)ls)[j];
        const float4 tot = mm2(red[0], ml);
        out[0] = tot.x;   // v0 = [1,0]  =>  v = row 0 of total product
        out[1] = tot.y;
    }
}

extern "C" void kernel_launch(void* const* d_in, const int* in_sizes, int n_in,
                              void* d_out, int out_size, void* d_ws, size_t ws_size,
                              hipStream_t stream) {
    const int*   tokens = (const int*)d_in[0];   // int32[4194304]
    const float* ls     = (const float*)d_in[1]; // float32[128*2*2]
    // d_in[2] ("final") is unused by the reference.
    float4* partial = (float4*)d_ws;             // 1024 * 16 B = 16 KB scratch

    hmm_partial<<<NBLK, TPB, 0, stream>>>(tokens, ls, partial);
    hmm_final<<<1, TPB, 0, stream>>>(partial, tokens, ls, (float*)d_out);
}